// ICUSTOMIntSoftmax_26852135535213
// MI455X (gfx1250) — compile-verified
//
#include <hip/hip_runtime.h>
#include <cmath>
#include <cstdint>
#include <climits>

// ---------------- configuration ----------------
#define ROWLEN 197
#define CHUNKS 7            // ceil(197/32)
#define WAVES_PER_BLOCK 8
#define BLOCK 256
#define SEGS 16

struct FitConsts {
    float lo[SEGS];   // float segment lower bounds
    float c0[SEGS];   // x^2 coefficient (highest power first, like np.polyfit)
    float c1[SEGS];
    float c2[SEGS];
};

// -------- CDNA5 async global->LDS path detection (device pass only) --------
#if defined(__HIP_DEVICE_COMPILE__) && defined(__gfx1250__)
#if __has_builtin(__builtin_amdgcn_global_load_async_to_lds_b32)
#define HAVE_ASYNC 1
#endif
#endif

#if defined(HAVE_ASYNC)
typedef __attribute__((address_space(1))) int* as1_i32p;
typedef __attribute__((address_space(1))) void* as1_vp;
typedef __attribute__((address_space(3))) int* as3_i32p;
typedef __attribute__((address_space(3))) void* as3_vp;

template <int N>
__device__ __forceinline__ void wait_async() {
#if __has_builtin(__builtin_amdgcn_s_wait_asynccnt)
    __builtin_amdgcn_s_wait_asynccnt(N);
#else
    if (N == 0) asm volatile("s_wait_asynccnt 0" ::: "memory");
    else        asm volatile("s_wait_asynccnt 7" ::: "memory");
#endif
    asm volatile("" ::: "memory");   // keep LDS reads after the wait
}

// Issue one row (197 floats) as 7 per-lane async b32 copies into this wave's
// LDS buffer (ASYNCcnt-tracked). Tail lanes clamp the *global* address (stay
// in-bounds) and are masked out later in compute.
__device__ __forceinline__ void issue_async_row(const float* rowp, float* dst, int lane) {
#pragma unroll
    for (int c = 0; c < CHUNKS; ++c) {
        int idx = c * 32 + lane;
        int g   = idx < ROWLEN ? idx : (ROWLEN - 1);
        as1_i32p gp = (as1_i32p)(as1_vp)(float*)(rowp + g);
        as3_i32p lp = (as3_i32p)(as3_vp)(dst + idx);
        __builtin_amdgcn_global_load_async_to_lds_b32(gp, lp, 0, 0);
    }
}
#endif  // HAVE_ASYNC

// ---------------- per-row compute (one wave per row) ----------------
// L2..L14: wave-uniform (readfirstlane'd) even entries of lo_int for the
// register levels of the binary search; odd entries gathered from LDS s_lo.
__device__ __forceinline__ void compute_row(const float v[CHUNKS], float rinv,
                                            int L2, int L4, int L6, int L8,
                                            int L10, int L12, int L14,
                                            const int* s_lo, const int4* s_coef,
                                            float* __restrict__ out,
                                            long long rowbase, int lane) {
    // pass 1: quantize (x * (1/s), floor) + wave max
    int xi[CHUNKS];
    int rmax = INT_MIN;
#pragma unroll
    for (int c = 0; c < CHUNKS; ++c) {
        int idx = c * 32 + lane;
        int q = (int)floorf(v[c] * rinv);     // fast quantize; rinv = precise 1/s
        if (idx >= ROWLEN) q = INT_MIN;       // masked lane: never the max
        xi[c] = q;
        rmax  = q > rmax ? q : rmax;
    }
#pragma unroll
    for (int off = 16; off > 0; off >>= 1) {  // wave32 butterfly max
        int o = __shfl_xor(rmax, off, 32);
        rmax = o > rmax ? o : rmax;
    }

    // pass 2: binary-search segment + int32 Horner + wave sum
    float ev[CHUNKS];
    float fsum = 0.0f;
#pragma unroll
    for (int c = 0; c < CHUNKS; ++c) {
        int idx = c * 32 + lane;
        int q = (int)((unsigned)xi[c] - (unsigned)rmax);   // wrapping sub (jnp int32)

        // seg = #{j in 1..15 : lo[j] <= q}  via 4-level binary search
        // (predicate lo[j] <= q is monotone since lo[] is sorted)
        int seg = (L8 <= q) ? 8 : 0;
        int tA  = (seg != 0) ? L12 : L4;
        seg += (tA <= q) ? 4 : 0;
        int u  = (seg & 4) ? L6  : L2;
        int w  = (seg & 4) ? L14 : L10;
        int tB = (seg & 8) ? w : u;
        seg += (tB <= q) ? 2 : 0;
        int tC = s_lo[seg + 1];                 // single LDS gather (odd entries)
        seg += (tC <= q) ? 1 : 0;

        int4 cc = s_coef[seg];                  // one ds_load_b128: {c0,c1,c2,pad}
        int r = cc.x;
        r = (int)((unsigned)r * (unsigned)q) + cc.y;   // wrapping int32 Horner
        r = (int)((unsigned)r * (unsigned)q) + cc.z;
        float e = (float)r;
        if (idx >= ROWLEN) e = 0.0f;
        ev[c] = e;
        fsum += e;
    }
#pragma unroll
    for (int off = 16; off > 0; off >>= 1)    // wave32 butterfly add
        fsum += __shfl_xor(fsum, off, 32);

    float esum   = fsum > 1.0f ? fsum : 1.0f;
    float factor = floorf(2147483647.0f / esum);
    factor = factor < 16777216.0f ? factor : 16777216.0f;  // 2^(32-8)
    // fold the exact /2^24 into the multiplier (power-of-two scale is exact,
    // so floorf(ev*g) == floorf((ev*factor) * 2^-24) bit-for-bit)
    float g = factor * (1.0f / 16777216.0f);

#pragma unroll
    for (int c = 0; c < CHUNKS; ++c) {
        int idx = c * 32 + lane;
        if (idx < ROWLEN)
            out[rowbase + idx] = floorf(ev[c] * g) * 0.0078125f;  // * out_s
    }
}

// ---------------- main kernel ----------------
__global__ __launch_bounds__(BLOCK) void intsoftmax_kernel(
    const float* __restrict__ x, const float* __restrict__ s_ptr,
    float* __restrict__ out, FitConsts fc, int rows, long long nelems) {
    __shared__ int  s_lo[SEGS];
    __shared__ int4 s_coef[SEGS];
#if defined(HAVE_ASYNC)
    __shared__ float s_buf[WAVES_PER_BLOCK * 2 * (CHUNKS * 32)];
#endif

    const float s    = *s_ptr;
    const float rinv = 1.0f / s;              // one precise division per thread

    // build runtime fixed-point LUTs (depend on scaling factor)
    if (threadIdx.x < SEGS) {
        int t = threadIdx.x;
        s_lo[t] = (int)floorf(fc.lo[t] / s);
        int c0 = (int)floorf(fc.c0[t] * (s * s) * 16777216.0f);
        int c1 = (int)floorf(fc.c1[t] * s * 16777216.0f);
        int c2 = (int)floorf(fc.c2[t] * 16777216.0f);
        s_coef[t] = make_int4(c0, c1, c2, 0);
    }
    __syncthreads();

    // even lo entries into SGPRs (wave-uniform) for the register search levels
    const int L2  = __builtin_amdgcn_readfirstlane(s_lo[2]);
    const int L4  = __builtin_amdgcn_readfirstlane(s_lo[4]);
    const int L6  = __builtin_amdgcn_readfirstlane(s_lo[6]);
    const int L8  = __builtin_amdgcn_readfirstlane(s_lo[8]);
    const int L10 = __builtin_amdgcn_readfirstlane(s_lo[10]);
    const int L12 = __builtin_amdgcn_readfirstlane(s_lo[12]);
    const int L14 = __builtin_amdgcn_readfirstlane(s_lo[14]);

    const int lane = threadIdx.x & 31;
    const int wid  = threadIdx.x >> 5;
    long long row        = (long long)blockIdx.x * WAVES_PER_BLOCK + wid;
    const long long strd = (long long)gridDim.x * WAVES_PER_BLOCK;

    if (blockIdx.x == 0 && threadIdx.x == 0)
        out[nelems] = 0.0078125f;             // second return value: out_s

#if defined(HAVE_ASYNC)
    float* bufA = &s_buf[(wid * 2 + 0) * (CHUNKS * 32)];
    float* bufB = &s_buf[(wid * 2 + 1) * (CHUNKS * 32)];
    if (row < rows) issue_async_row(x + row * ROWLEN, bufA, lane);
    for (long long r = row; r < rows; r += strd) {
        long long nr = r + strd;
        bool pre = nr < (long long)rows;
        if (pre) issue_async_row(x + nr * ROWLEN, bufB, lane);   // prefetch next row
        if (pre) wait_async<CHUNKS>();        // 7 newest in flight -> bufA complete
        else     wait_async<0>();
        float v[CHUNKS];
#pragma unroll
        for (int c = 0; c < CHUNKS; ++c) v[c] = bufA[c * 32 + lane];
        compute_row(v, rinv, L2, L4, L6, L8, L10, L12, L14,
                    s_lo, s_coef, out, r * ROWLEN, lane);
        float* t = bufA; bufA = bufB; bufB = t;
    }
#else
    for (long long r = row; r < rows; r += strd) {
        const float* rp = x + r * ROWLEN;
        float v[CHUNKS];
#pragma unroll
        for (int c = 0; c < CHUNKS; ++c) {
            int idx = c * 32 + lane;
            int g   = idx < ROWLEN ? idx : (ROWLEN - 1);
            v[c] = rp[g];
        }
        compute_row(v, rinv, L2, L4, L6, L8, L10, L12, L14,
                    s_lo, s_coef, out, r * ROWLEN, lane);
    }
#endif
}

// ---------------- host: reproduce np.polyfit of exp(x) on [-10,0] ----------------
static FitConsts make_fit() {
    FitConsts fc;
    const int NPTS = 10000;
    float bounds[SEGS + 1];
    for (int i = 0; i <= SEGS; ++i)
        bounds[i] = (float)(-10.0 + 10.0 * (double)i / (double)SEGS);

    double S0[SEGS] = {}, S1[SEGS] = {}, S2[SEGS] = {}, S3[SEGS] = {}, S4[SEGS] = {};
    double T0[SEGS] = {}, T1[SEGS] = {}, T2[SEGS] = {};

    for (int i = 0; i < NPTS; ++i) {
        float xf = (float)(-10.0 + 10.0 * (double)i / (double)(NPTS - 1));
        float yf = (float)exp((double)xf);
        double xd = (double)xf, yd = (double)yf;
        for (int sgi = 0; sgi < SEGS; ++sgi) {
            if (xf >= bounds[sgi] && xf <= bounds[sgi + 1]) {
                double x2 = xd * xd;
                S0[sgi] += 1.0;       S1[sgi] += xd;
                S2[sgi] += x2;        S3[sgi] += x2 * xd;
                S4[sgi] += x2 * x2;
                T0[sgi] += yd;        T1[sgi] += xd * yd;
                T2[sgi] += x2 * yd;
            }
        }
    }

    for (int sgi = 0; sgi < SEGS; ++sgi) {
        double A[3][3] = {{S4[sgi], S3[sgi], S2[sgi]},
                          {S3[sgi], S2[sgi], S1[sgi]},
                          {S2[sgi], S1[sgi], S0[sgi]}};
        double b[3] = {T2[sgi], T1[sgi], T0[sgi]};
        // Gaussian elimination w/ partial pivoting
        for (int k = 0; k < 3; ++k) {
            int p = k;
            for (int i2 = k + 1; i2 < 3; ++i2)
                if (fabs(A[i2][k]) > fabs(A[p][k])) p = i2;
            if (p != k) {
                for (int j = 0; j < 3; ++j) { double t = A[k][j]; A[k][j] = A[p][j]; A[p][j] = t; }
                double t = b[k]; b[k] = b[p]; b[p] = t;
            }
            for (int i2 = k + 1; i2 < 3; ++i2) {
                double f = A[i2][k] / A[k][k];
                for (int j = k; j < 3; ++j) A[i2][j] -= f * A[k][j];
                b[i2] -= f * b[k];
            }
        }
        double c[3];
        for (int k = 2; k >= 0; --k) {
            double v = b[k];
            for (int j = k + 1; j < 3; ++j) v -= A[k][j] * c[j];
            c[k] = v / A[k][k];
        }
        fc.lo[sgi] = bounds[sgi];
        fc.c0[sgi] = (float)c[0];
        fc.c1[sgi] = (float)c[1];
        fc.c2[sgi] = (float)c[2];
    }
    return fc;
}

// ---------------- entry point ----------------
extern "C" void kernel_launch(void* const* d_in, const int* in_sizes, int n_in,
                              void* d_out, int out_size, void* d_ws, size_t ws_size,
                              hipStream_t stream) {
    (void)n_in; (void)d_ws; (void)ws_size; (void)out_size;
    const float* x  = (const float*)d_in[0];
    const float* sf = (const float*)d_in[1];
    float* out = (float*)d_out;

    int n    = in_sizes[0];
    int rows = n / ROWLEN;

    FitConsts fc = make_fit();

    int blocks = 2048;  // persistent-ish: ~9 rows per wave keeps the async pipeline busy
    int maxb = (rows + WAVES_PER_BLOCK - 1) / WAVES_PER_BLOCK;
    if (blocks > maxb) blocks = maxb;

    intsoftmax_kernel<<<blocks, BLOCK, 0, stream>>>(x, sf, out, fc, rows, (long long)n);
}